// VAE_20143396618969
// MI455X (gfx1250) — compile-verified
//
#include <hip/hip_runtime.h>
#include <math.h>

typedef __attribute__((ext_vector_type(2))) float v2f;
typedef __attribute__((ext_vector_type(8))) float v8f;

#define NN 8192
#define NE 262144

__device__ __forceinline__ v8f wmma4(v2f a, v2f b, v8f c) {
    // D = A(16x4,f32) * B(4x16,f32) + C(16x16,f32), full fp32 precision
    return __builtin_amdgcn_wmma_f32_16x16x4_f32(
        /*neg_a=*/false, a, /*neg_b=*/false, b,
        /*c_mod=*/(short)0, c, /*reuse_a=*/false, /*reuse_b=*/false);
}

// ---------------- degree / normalization ----------------
__global__ void deg_init_k(float* deg) {
    int i = blockIdx.x * blockDim.x + threadIdx.x;
    if (i < NN) deg[i] = 1.0f;                       // self-loop weight
}
__global__ void deg_accum_k(const int* __restrict__ ei, const float* __restrict__ w,
                            float* __restrict__ deg) {
    int e = blockIdx.x * blockDim.x + threadIdx.x;
    if (e < NE) atomicAdd(&deg[ei[NE + e]], w[e]);   // col = ei[E + e]
}
__global__ void dinv_k(const float* __restrict__ deg, float* __restrict__ dinv) {
    int i = blockIdx.x * blockDim.x + threadIdx.x;
    if (i < NN) dinv[i] = rsqrtf(deg[i]);
}
__global__ void norm_k(const int* __restrict__ ei, const float* __restrict__ w,
                       const float* __restrict__ dinv, float* __restrict__ normw) {
    int e = blockIdx.x * blockDim.x + threadIdx.x;
    if (e < NE) normw[e] = dinv[ei[e]] * w[e] * dinv[ei[NE + e]];
}

// ---------------- dense GEMM: C[M,Nc] = A[M,K] @ B[K,Nc], f32 WMMA ----------
// one wave -> 16x64 strip (4 accumulators); M%16==0, K%4==0, Nc%64==0
__global__ __launch_bounds__(256) void wmma_gemm_f32(
    const float* __restrict__ A, const float* __restrict__ B, float* __restrict__ C,
    int M, int K, int Nc)
{
    const int wave = threadIdx.x >> 5;
    const int lane = threadIdx.x & 31;
    const int ctiles = Nc >> 6;
    const int nTasks = (M >> 4) * ctiles;
    int task = blockIdx.x * 8 + wave;                // wave-uniform
    if (task >= nTasks) return;
    int rt = task / ctiles, ct = task - rt * ctiles;
    int r0 = rt << 4, c0 = ct << 6;
    int m = lane & 15, hi = lane >> 4;

    const float* arow = A + (size_t)(r0 + m) * K + 2 * hi;
    v8f acc0 = {}, acc1 = {}, acc2 = {}, acc3 = {};
    for (int k0 = 0; k0 < K; k0 += 4) {
        v2f a = *(const v2f*)(arow + k0);
        const float* bp = B + (size_t)(k0 + 2 * hi) * Nc + c0 + m;
        v2f b0, b1, b2, b3;
        b0.x = bp[0];  b0.y = bp[Nc];
        b1.x = bp[16]; b1.y = bp[Nc + 16];
        b2.x = bp[32]; b2.y = bp[Nc + 32];
        b3.x = bp[48]; b3.y = bp[Nc + 48];
        acc0 = wmma4(a, b0, acc0);
        acc1 = wmma4(a, b1, acc1);
        acc2 = wmma4(a, b2, acc2);
        acc3 = wmma4(a, b3, acc3);
    }
    float* crow = C + (size_t)c0 + m;
    #pragma unroll
    for (int v = 0; v < 8; ++v) {
        size_t ro = (size_t)(r0 + v + 8 * hi) * Nc;
        crow[ro]      = acc0[v];
        crow[ro + 16] = acc1[v];
        crow[ro + 32] = acc2[v];
        crow[ro + 48] = acc3[v];
    }
}

// ---------------- SpMM pieces ----------------
__global__ void selfloop_init_k(const float* __restrict__ dinv, const float* __restrict__ XW,
                                float* __restrict__ agg, int D) {
    int idx = blockIdx.x * blockDim.x + threadIdx.x;   // N*D threads
    int i = idx / D;
    float di = dinv[i];
    agg[idx] = di * di * XW[idx];
}
__global__ void spmm_scatter_k(const int* __restrict__ ei, const float* __restrict__ normw,
                               const float* __restrict__ XW, float* __restrict__ agg, int D) {
    long long idx = (long long)blockIdx.x * blockDim.x + threadIdx.x;  // E*D threads
    int e = (int)(idx / D);
    int f = (int)(idx - (long long)e * D);
    int r = ei[e], c = ei[NE + e];
    atomicAdd(&agg[(size_t)c * D + f], normw[e] * XW[(size_t)r * D + f]);
}
__global__ void bias_relu_k(const float* __restrict__ agg, const float* __restrict__ b,
                            float* __restrict__ out, int D) {
    int idx = blockIdx.x * blockDim.x + threadIdx.x;   // N*D threads
    int f = idx % D;
    out[idx] = fmaxf(agg[idx] + b[f], 0.0f);
}

// ---------------- reparameterize ----------------
__global__ void z_k(const float* __restrict__ noise, const float* __restrict__ s,
                    const float* __restrict__ m, float* __restrict__ z) {
    int idx = blockIdx.x * blockDim.x + threadIdx.x;   // N*64 threads
    z[idx] = noise[idx] * __expf(s[idx]) + m[idx];
}

// ---------------- out = triu(sigmoid(z @ z^T), 1), nontemporal 256MB write ---
__global__ __launch_bounds__(256) void zzt_sigmoid_triu_k(
    const float* __restrict__ Z, float* __restrict__ out)
{
    const int wave = threadIdx.x >> 5;
    const int lane = threadIdx.x & 31;
    const int CT = NN >> 6;                            // 128 col strips
    int task = blockIdx.x * 8 + wave;                  // 512*128 = 65536 tasks
    if (task >= (NN >> 4) * CT) return;
    int rt = task / CT, ct = task - rt * CT;
    int r0 = rt << 4, c0 = ct << 6;
    int m = lane & 15, hi = lane >> 4;

    float* base = out + (size_t)c0 + m;
    if (c0 + 63 <= r0) {                               // tile entirely j<=i: zeros
        #pragma unroll
        for (int v = 0; v < 8; ++v) {
            size_t ro = (size_t)(r0 + v + 8 * hi) * NN;
            __builtin_nontemporal_store(0.0f, base + ro);
            __builtin_nontemporal_store(0.0f, base + ro + 16);
            __builtin_nontemporal_store(0.0f, base + ro + 32);
            __builtin_nontemporal_store(0.0f, base + ro + 48);
        }
        return;
    }

    const float* arow = Z + (size_t)(r0 + m) * 64 + 2 * hi;
    const float* zc0 = Z + (size_t)(c0 + m) * 64 + 2 * hi;
    v8f acc0 = {}, acc1 = {}, acc2 = {}, acc3 = {};
    #pragma unroll
    for (int k0 = 0; k0 < 64; k0 += 4) {
        v2f a  = *(const v2f*)(arow + k0);
        v2f b0 = *(const v2f*)(zc0 + k0);              // Z^T[k][n] = Z[n][k]
        v2f b1 = *(const v2f*)(zc0 + 16 * 64 + k0);
        v2f b2 = *(const v2f*)(zc0 + 32 * 64 + k0);
        v2f b3 = *(const v2f*)(zc0 + 48 * 64 + k0);
        acc0 = wmma4(a, b0, acc0);
        acc1 = wmma4(a, b1, acc1);
        acc2 = wmma4(a, b2, acc2);
        acc3 = wmma4(a, b3, acc3);
    }
    #pragma unroll
    for (int v = 0; v < 8; ++v) {
        int i = r0 + v + 8 * hi;
        size_t ro = (size_t)i * NN;
        int j = c0 + m;
        float o0 = (j      > i) ? 1.0f / (1.0f + __expf(-acc0[v])) : 0.0f;
        float o1 = (j + 16 > i) ? 1.0f / (1.0f + __expf(-acc1[v])) : 0.0f;
        float o2 = (j + 32 > i) ? 1.0f / (1.0f + __expf(-acc2[v])) : 0.0f;
        float o3 = (j + 48 > i) ? 1.0f / (1.0f + __expf(-acc3[v])) : 0.0f;
        __builtin_nontemporal_store(o0, base + ro);
        __builtin_nontemporal_store(o1, base + ro + 16);
        __builtin_nontemporal_store(o2, base + ro + 32);
        __builtin_nontemporal_store(o3, base + ro + 48);
    }
}

// ---------------- host orchestration ----------------
static void run_layer(const float* in, int K, const float* W, const float* b, int D,
                      const int* ei, const float* normw, const float* dinv,
                      float* bufA, float* bufB, float* outp, hipStream_t s)
{
    int tiles = (NN / 16) * (D / 64);
    wmma_gemm_f32<<<(tiles + 7) / 8, 256, 0, s>>>(in, W, bufA, NN, K, D);
    int nd = NN * D;
    selfloop_init_k<<<nd / 256, 256, 0, s>>>(dinv, bufA, bufB, D);
    unsigned sblocks = (unsigned)(((long long)NE * D) / 256);
    spmm_scatter_k<<<sblocks, 256, 0, s>>>(ei, normw, bufA, bufB, D);
    bias_relu_k<<<nd / 256, 256, 0, s>>>(bufB, b, outp, D);
}

extern "C" void kernel_launch(void* const* d_in, const int* in_sizes, int n_in,
                              void* d_out, int out_size, void* d_ws, size_t ws_size,
                              hipStream_t stream)
{
    const float* x     = (const float*)d_in[0];
    const int*   ei    = (const int*)d_in[1];
    const float* ew    = (const float*)d_in[2];
    const float* noise = (const float*)d_in[3];
    const float* W1 = (const float*)d_in[4];
    const float* b1 = (const float*)d_in[5];
    const float* W2 = (const float*)d_in[6];
    const float* b2 = (const float*)d_in[7];
    const float* Wm = (const float*)d_in[8];
    const float* bm = (const float*)d_in[9];
    const float* Wsp = (const float*)d_in[10];
    const float* bsp = (const float*)d_in[11];

    float* ws    = (float*)d_ws;
    float* deg   = ws;                    // 8192
    float* dinv  = deg   + NN;            // 8192
    float* normw = dinv  + NN;            // E
    float* bufA  = normw + NE;            // 8192*256
    float* bufB  = bufA  + NN * 256;      // 8192*256
    float* h     = bufB  + NN * 256;      // 8192*256
    float* mbuf  = h     + NN * 256;      // 8192*64
    float* zbuf  = mbuf  + NN * 64;       // 8192*64   (~30.5 MB total)

    // normalization, shared by all layers
    deg_init_k <<<NN / 256, 256, 0, stream>>>(deg);
    deg_accum_k<<<NE / 256, 256, 0, stream>>>(ei, ew, deg);
    dinv_k     <<<NN / 256, 256, 0, stream>>>(deg, dinv);
    norm_k     <<<NE / 256, 256, 0, stream>>>(ei, ew, dinv, normw);

    // h = relu(gcn(x, W1, b1)); h = relu(gcn(h, W2, b2))
    run_layer(x, 512, W1, b1, 256, ei, normw, dinv, bufA, bufB, h, stream);
    run_layer(h, 256, W2, b2, 256, ei, normw, dinv, bufA, bufB, h, stream);
    // mean / log_std heads (64-wide)
    run_layer(h, 256, Wm,  bm,  64, ei, normw, dinv, bufA, bufB, mbuf, stream);
    run_layer(h, 256, Wsp, bsp, 64, ei, normw, dinv, bufA, bufB, zbuf, stream); // zbuf=log_std
    // z = noise * exp(log_std) + mean (in place over zbuf)
    z_k<<<NN * 64 / 256, 256, 0, stream>>>(noise, zbuf, mbuf, zbuf);
    // out = triu(sigmoid(z @ z^T), 1)
    zzt_sigmoid_triu_k<<<(NN / 16) * (NN / 64) / 8, 256, 0, stream>>>(zbuf, (float*)d_out);
}